// GNN_Module_37074157699470
// MI455X (gfx1250) — compile-verified
//
#include <hip/hip_runtime.h>
#include <cstdint>

// Problem constants (match reference)
#define BB   256
#define NPTS 325
#define DD   64
#define TI   21            // ceil(NPTS/16)
#define EPSV 1e-5f

typedef float v2f __attribute__((ext_vector_type(2)));
typedef float v8f __attribute__((ext_vector_type(8)));

// ---------------------------------------------------------------------------
// Kernel A: hw[b,n,:] = h[b,n,:] @ W   (per batch: (N x 64) x (64 x 64))
// 4 waves per block; wave w owns the 16x16 tile (rows n0.., cols 16w..),
// 16 steps of V_WMMA_F32_16X16X4_F32 over K=64.
// ---------------------------------------------------------------------------
__global__ __launch_bounds__(128)
void gcn_linear(const float* __restrict__ h,
                const float* __restrict__ Wl,
                float* __restrict__ hw)
{
    __shared__ float hs[16 * 68];   // padded stride 68 -> conflict-free A reads

    const int tid = threadIdx.x;
    const int b   = blockIdx.x / TI;
    const int n0  = (blockIdx.x % TI) * 16;

    for (int e = tid; e < 16 * 64; e += 128) {
        const int r = e >> 6, c = e & 63;
        const int n = n0 + r;
        hs[r * 68 + c] = (n < NPTS) ? h[((size_t)b * NPTS + n) * DD + c] : 0.f;
    }
    __syncthreads();

    const int lane = tid & 31;
    const int wid  = tid >> 5;
    const int d0   = wid * 16;
    const int m    = lane & 15;
    const int kb   = (lane >> 4) * 2;
    const int dcol = d0 + m;

    v8f acc = {0.f, 0.f, 0.f, 0.f, 0.f, 0.f, 0.f, 0.f};

#pragma unroll
    for (int s = 0; s < 16; ++s) {
        const int k = 4 * s + kb;
        v2f a, bf;
        a.x  = hs[m * 68 + k];
        a.y  = hs[m * 68 + k + 1];
        bf.x = Wl[(k)     * DD + dcol];
        bf.y = Wl[(k + 1) * DD + dcol];
        acc = __builtin_amdgcn_wmma_f32_16x16x4_f32(false, a, false, bf,
                                                    (short)0, acc, false, false);
    }

#pragma unroll
    for (int r = 0; r < 8; ++r) {
        const int row = r + 8 * (lane >> 4);
        const int n   = n0 + row;
        if (n < NPTS) hw[((size_t)b * NPTS + n) * DD + dcol] = acc[r];
    }
}

// ---------------------------------------------------------------------------
// Kernel B: out = ReLU(LN(adj^T @ hw + bias + h)) per batch, double-buffered
// through LDS with GLOBAL_LOAD_ASYNC_TO_LDS (ASYNCcnt) on interior chunks.
// ---------------------------------------------------------------------------
__global__ __launch_bounds__(128)
void gcn_agg_ln(const float* __restrict__ adj,
                const float* __restrict__ hw,
                const float* __restrict__ hin,
                const float* __restrict__ bias,
                const float* __restrict__ gamma,
                const float* __restrict__ beta,
                float* __restrict__ hout)
{
    __shared__ float adjS[2][32 * 16];   // adj^T tile: [jj][ii]
    __shared__ float hwS[2][32 * 80];    // hw tile [jj][d], stride 80
    __shared__ float yS[16 * 65];        // LN staging, stride 65
    __shared__ float muS[16], rsS[16];

    const int tid = threadIdx.x;
    const int b   = blockIdx.x / TI;
    const int i0  = (blockIdx.x % TI) * 16;

    const int lane = tid & 31;
    const int wid  = tid >> 5;
    const int d0   = wid * 16;
    const int m    = lane & 15;
    const int kb   = (lane >> 4) * 2;
    const int dcol = d0 + m;

    const float* adjB = adj + (size_t)b * NPTS * NPTS;
    const float* hwB  = hw  + (size_t)b * NPTS * DD;

    // Guarded synchronous staging (edges / first chunk)
    auto load_sync = [&](int j0, int buf) {
        {
            const int jj  = tid >> 2;
            const int ii0 = (tid & 3) * 4;
            const int j   = j0 + jj;
#pragma unroll
            for (int q = 0; q < 4; ++q) {
                const int i = i0 + ii0 + q;
                float v = 0.f;
                if (j < NPTS && i < NPTS) v = adjB[(size_t)j * NPTS + i];
                adjS[buf][jj * 16 + ii0 + q] = v;
            }
        }
        for (int e = tid; e < 32 * 64; e += 128) {
            const int jj = e >> 6, c = e & 63;
            const int j  = j0 + jj;
            hwS[buf][jj * 80 + c] = (j < NPTS) ? hwB[(size_t)j * DD + c] : 0.f;
        }
    };

    // Async staging via CDNA5 GLOBAL_LOAD_ASYNC_TO_LDS (interior chunks only:
    // all rows/cols in-range, per-lane b128 addresses 16B-aligned for hw)
    auto load_async = [&](int j0, int buf) {
        {
            const int jj  = tid >> 2;
            const int ii0 = (tid & 3) * 4;
            const float* g = adjB + (size_t)(j0 + jj) * NPTS + i0 + ii0;
            uint32_t l = (uint32_t)(uintptr_t)&adjS[buf][jj * 16 + ii0];
#pragma unroll
            for (int q = 0; q < 4; ++q) {
                asm volatile("global_load_async_to_lds_b32 %0, %1, off"
                             :: "v"(l + 4u * q),
                                "v"((uint64_t)(uintptr_t)(g + q))
                             : "memory");
            }
        }
#pragma unroll
        for (int rr = 0; rr < 4; ++rr) {
            const int base = rr * 512 + tid * 4;   // float index in 32x64 tile
            const int jj   = base >> 6;
            const int c4   = base & 63;
            const float* g = hwB + (size_t)(j0 + jj) * DD + c4;
            uint32_t l = (uint32_t)(uintptr_t)&hwS[buf][jj * 80 + c4];
            asm volatile("global_load_async_to_lds_b128 %0, %1, off"
                         :: "v"(l), "v"((uint64_t)(uintptr_t)g)
                         : "memory");
        }
    };

    v8f acc = {0.f, 0.f, 0.f, 0.f, 0.f, 0.f, 0.f, 0.f};

    const int  NCH      = (NPTS + 31) / 32;     // 11
    const bool edgeTile = (i0 + 16 > NPTS);     // last row-tile: adj cols OOB

    load_sync(0, 0);
    __syncthreads();

    for (int ch = 0; ch < NCH; ++ch) {
        const int cur = ch & 1;

        if (ch + 1 < NCH) {
            const int  j0n  = (ch + 1) * 32;
            const bool fast = !edgeTile && (j0n + 32 <= NPTS);
            if (fast) load_async(j0n, 1 - cur);
            else      load_sync (j0n, 1 - cur);
        }

        const float* aS = adjS[cur];
        const float* hS = hwS[cur];
#pragma unroll
        for (int s = 0; s < 8; ++s) {
            const int k = 4 * s + kb;
            v2f a, bf;
            a.x  = aS[(k)     * 16 + m];
            a.y  = aS[(k + 1) * 16 + m];
            bf.x = hS[(k)     * 80 + dcol];
            bf.y = hS[(k + 1) * 80 + dcol];
            acc = __builtin_amdgcn_wmma_f32_16x16x4_f32(false, a, false, bf,
                                                        (short)0, acc, false, false);
        }

        // Drain this wave's async loads, then make next buffer visible to all
        asm volatile("s_wait_asynccnt 0" ::: "memory");
        __syncthreads();
    }

    // bias + residual -> yS
    {
        const float bv = bias[dcol];
#pragma unroll
        for (int r = 0; r < 8; ++r) {
            const int row = r + 8 * (lane >> 4);
            const int n   = i0 + row;
            const float res =
                (n < NPTS) ? hin[((size_t)b * NPTS + n) * DD + dcol] : 0.f;
            yS[row * 65 + dcol] = acc[r] + bv + res;
        }
    }
    __syncthreads();

    // Row-wise mean / rstd over D=64 (16 lanes, stride-65 rotated banks)
    if (tid < 16) {
        float s = 0.f;
#pragma unroll
        for (int c = 0; c < 64; ++c) s += yS[tid * 65 + c];
        const float mu = s * (1.f / 64.f);
        float v = 0.f;
#pragma unroll
        for (int c = 0; c < 64; ++c) {
            const float d = yS[tid * 65 + c] - mu;
            v += d * d;
        }
        muS[tid] = mu;
        rsS[tid] = rsqrtf(v * (1.f / 64.f) + EPSV);
    }
    __syncthreads();

    // Normalize + affine + ReLU + store
    for (int e = tid; e < 16 * 64; e += 128) {
        const int row = e >> 6, c = e & 63;
        const int n   = i0 + row;
        if (n < NPTS) {
            float val = (yS[row * 65 + c] - muS[row]) * rsS[row] * gamma[c] + beta[c];
            hout[((size_t)b * NPTS + n) * DD + c] = fmaxf(val, 0.f);
        }
    }
}

// ---------------------------------------------------------------------------
extern "C" void kernel_launch(void* const* d_in, const int* in_sizes, int n_in,
                              void* d_out, int out_size, void* d_ws, size_t ws_size,
                              hipStream_t stream)
{
    const float* X   = (const float*)d_in[0];
    const float* adj = (const float*)d_in[1];
    const float* W   = (const float*)d_in[2];
    const float* bi  = (const float*)d_in[3];
    const float* ga  = (const float*)d_in[4];
    const float* be  = (const float*)d_in[5];
    float*       out = (float*)d_out;
    float*       ws  = (float*)d_ws;

    const size_t BND = (size_t)BB * NPTS * DD;   // 5,324,800 floats (~21.3 MB)
    float* hwb = ws;                // hW scratch
    float* h0  = ws + BND;          // layer-1 output
    float* h1  = ws + 2 * BND;      // layer-2 output   (needs ~64 MB of d_ws)

    const dim3 grid(BB * TI);
    const dim3 blk(128);

    const float* hin = X;
    float* houts[3] = { h0, h1, out };
    for (int l = 0; l < 3; ++l) {
        gcn_linear<<<grid, blk, 0, stream>>>(hin, W + (size_t)l * DD * DD, hwb);
        gcn_agg_ln<<<grid, blk, 0, stream>>>(adj, hwb, hin,
                                             bi + l * DD, ga + l * DD, be + l * DD,
                                             houts[l]);
        hin = houts[l];
    }
}